// SelectReLU_64905545777512
// MI455X (gfx1250) — compile-verified
//
#include <hip/hip_runtime.h>
#include <stdint.h>

// SelectReLU top-k threshold kernel for MI455X (gfx1250).
// One workgroup per row (16384 fp32). Row staged HBM->LDS with CDNA5
// global_load_async_to_lds_b128 (ASYNCcnt path), k-th largest found with a
// 3-round radix select (11+10+10 bits) over the monotone uint bit patterns of
// the (implicitly ReLU'd) non-negative floats, zeros excluded to avoid LDS
// same-address atomic serialization. Predicated 128-bit writeback.

#define COLS 16384
#define KSEL 1638u      // floor(0.1 * 16384)
#define BS   512        // 16 waves of 32

__device__ __forceinline__ unsigned relu_bits(unsigned u) {
    return (u >> 31) ? 0u : u;   // negative (incl. -0) -> 0
}

__global__ __launch_bounds__(BS)
void selectrelu_topk_kernel(const float* __restrict__ x,
                            float* __restrict__ out,
                            int rows) {
    extern __shared__ __align__(16) unsigned smem[];
    unsigned* sbits = smem;                    // COLS words (64 KB): raw row bits
    unsigned* hist  = smem + COLS;             // up to 2048 bins (8 KB)
    unsigned* gsum  = smem + COLS + 2048;      // up to 64 group sums
    unsigned* sbc   = smem + COLS + 2048 + 64; // [0]=sel bin [1]=rem k [2]=zero-thresh flag

    const int row = blockIdx.x;
    if (row >= rows) return;
    const int tid = threadIdx.x;

    const uint64_t gbase = (uint64_t)(uintptr_t)(x + (size_t)row * COLS);
    const unsigned sbase = (unsigned)(uintptr_t)(void*)sbits; // LDS byte offset

    // ---- Phase 1: async copy row (64 KB) HBM -> LDS, 16 B per lane ----
    #pragma unroll
    for (int it = 0; it < COLS / (BS * 4); ++it) {
        unsigned goff  = (unsigned)(it * BS + tid) * 16u;
        unsigned laddr = sbase + goff;
        asm volatile("global_load_async_to_lds_b128 %0, %1, %2"
                     :: "v"(laddr), "v"(goff), "s"(gbase)
                     : "memory");
    }
    if (tid == 0) sbc[2] = 0u;
    asm volatile("s_wait_asynccnt 0" ::: "memory");
    __syncthreads();

    // ---- Phase 2: radix select k-th largest: 11 + 10 + 10 bits ----
    // Zeros are excluded from all rounds; if #positives < k the threshold is 0.
    unsigned prefix = 0u, mask = 0u, kk = KSEL;
    #pragma unroll
    for (int r = 0; r < 3; ++r) {
        const int      shift = (r == 0) ? 20 : (r == 1) ? 10 : 0;
        const unsigned im    = (r == 0) ? 0x7FFu : 0x3FFu;
        const int      nb    = (r == 0) ? 2048 : 1024;
        const int      G     = nb / 32;

        for (int i = tid; i < nb; i += BS) hist[i] = 0u;
        __syncthreads();

        for (int i = tid; i < COLS; i += BS) {
            unsigned u = relu_bits(sbits[i]);
            if (u != 0u && (u & mask) == prefix)
                atomicAdd(&hist[(u >> shift) & im], 1u);
        }
        __syncthreads();

        // Parallel group sums (32 bins per group), then short serial scan.
        if (tid < G) {
            unsigned s = 0u;
            #pragma unroll
            for (int j = 0; j < 32; ++j) s += hist[tid * 32 + j];
            gsum[tid] = s;
        }
        __syncthreads();

        if (tid == 0) {
            unsigned cum = 0u, sel = 0u, kkn = kk;
            int g = -1;
            for (int gg = G - 1; gg >= 0; --gg) {
                unsigned c = gsum[gg];
                if (cum + c >= kk) { g = gg; break; }
                cum += c;
            }
            if (g < 0) {
                sbc[2] = 1u;  // fewer than k positive elements: threshold = 0
            } else {
                for (int b = g * 32 + 31; b >= g * 32; --b) {
                    unsigned c = hist[b];
                    if (cum + c >= kk) { sel = (unsigned)b; kkn = kk - cum; break; }
                    cum += c;
                }
            }
            sbc[0] = sel;
            sbc[1] = kkn;
        }
        __syncthreads();

        prefix |= sbc[0] << shift;
        mask   |= im << shift;
        kk      = sbc[1];
    }

    __syncthreads();
    const unsigned tb = sbc[2] ? 0u : prefix; // bit pattern of k-th largest

    // ---- Phase 3: predicated writeback, 128-bit LDS reads + global stores ----
    float* outr = out + (size_t)row * COLS;
    for (int i = tid * 4; i < COLS; i += BS * 4) {
        uint4 u = *reinterpret_cast<const uint4*>(&sbits[i]);
        unsigned ax = relu_bits(u.x), ay = relu_bits(u.y);
        unsigned az = relu_bits(u.z), aw = relu_bits(u.w);
        float4 o;
        o.x = (ax >= tb) ? __uint_as_float(ax) : 0.0f;
        o.y = (ay >= tb) ? __uint_as_float(ay) : 0.0f;
        o.z = (az >= tb) ? __uint_as_float(az) : 0.0f;
        o.w = (aw >= tb) ? __uint_as_float(aw) : 0.0f;
        *reinterpret_cast<float4*>(&outr[i]) = o;
    }
}

extern "C" void kernel_launch(void* const* d_in, const int* in_sizes, int n_in,
                              void* d_out, int out_size, void* d_ws, size_t ws_size,
                              hipStream_t stream) {
    (void)n_in; (void)d_ws; (void)ws_size; (void)in_sizes;
    const float* x = (const float*)d_in[0];
    float* out = (float*)d_out;
    const int rows = out_size / COLS;  // 2048
    const size_t shmem = (size_t)(COLS + 2048 + 64 + 8) * sizeof(unsigned); // ~74 KB
    selectrelu_topk_kernel<<<rows, BS, shmem, stream>>>(x, out, rows);
}